// StructureModule_42580305772628
// MI455X (gfx1250) — compile-verified
//
#include <hip/hip_runtime.h>

// ---------------------------------------------------------------------------
// CDNA5 (gfx1250) structure-module implementation.
// All dense math runs through V_WMMA_F32_16X16X4_F32 (exact fp32 matrix op).
// ---------------------------------------------------------------------------

typedef float v2f __attribute__((ext_vector_type(2)));
typedef float v4f __attribute__((ext_vector_type(4)));
typedef float v8f __attribute__((ext_vector_type(8)));

#define NRES 512
#define CS   384
#define CZ   128
#define CI   16
#define CR   128
#define HH   12
#define PQ   4
#define PV   8
#define CATW 2112   // H*(CI + CZ + PV*4)
#define OUTW 415    // CS + 3 + 14 + 14

// ---------------------------------------------------------------------------
// Generic batched WMMA GEMM:  C = act(A @ B + bias) [+ add]
//   A: [M,K] row-major, lda;  B: [K,N] (or [N,K] if transB), ldb
//   batch via blockIdx.z (element strides sA/sB/sC).
//   Block = 128 threads = 4 waves; wave w computes C tile (m0, (4*bx+w)*16),
//   K in steps of 4 via V_WMMA_F32_16X16X4_F32.
//   Uniform fast path (branchless b64 loads) when the tile is fully in range
//   and K % 4 == 0; guarded path only for edge tiles.
// ---------------------------------------------------------------------------
__global__ void __launch_bounds__(128) gemm_wmma_f32(
                              const float* __restrict__ A,
                              const float* __restrict__ B,
                              const float* __restrict__ bias,
                              const float* __restrict__ add,
                              float* __restrict__ C,
                              int M, int N, int K,
                              int lda, int ldb, int ldc,
                              long long sA, long long sB, long long sC,
                              int transB, int reluA, int reluOut)
{
    const int wave = threadIdx.x >> 5;
    const int lane = threadIdx.x & 31;
    const int n0   = (blockIdx.x * 4 + wave) * 16;
    const int m0   = blockIdx.y * 16;
    if (n0 >= N) return;                       // wave-uniform

    const long long b = blockIdx.z;
    A += b * sA;  B += b * sB;  C += b * sC;
    if (add) add += b * sC;

    const int half = lane >> 4;       // selects K pair within a 4-step
    const int lm   = lane & 15;       // M row (for A) / N col (for B,C)
    const int row  = m0 + lm;
    const int col  = n0 + lm;

    v8f acc = {};
    const bool edge = (m0 + 16 > M) || (n0 + 16 > N) || (K & 3);

    if (!edge) {
        const float* Ap = A + (long long)row * lda + 2 * half;
        if (!transB) {
            const float* Bp = B + (long long)(2 * half) * ldb + col;
            if (!reluA) {
                for (int k0 = 0; k0 < K; k0 += 4) {
                    v2f av = *(const v2f*)(Ap + k0);
                    v2f bv;
                    bv.x = Bp[(long long)k0 * ldb];
                    bv.y = Bp[(long long)(k0 + 1) * ldb];
                    acc = __builtin_amdgcn_wmma_f32_16x16x4_f32(
                        false, av, false, bv, (short)0, acc, false, false);
                }
            } else {
                for (int k0 = 0; k0 < K; k0 += 4) {
                    v2f av = *(const v2f*)(Ap + k0);
                    av.x = fmaxf(av.x, 0.f); av.y = fmaxf(av.y, 0.f);
                    v2f bv;
                    bv.x = Bp[(long long)k0 * ldb];
                    bv.y = Bp[(long long)(k0 + 1) * ldb];
                    acc = __builtin_amdgcn_wmma_f32_16x16x4_f32(
                        false, av, false, bv, (short)0, acc, false, false);
                }
            }
        } else {
            const float* Bp = B + (long long)col * ldb + 2 * half;
            for (int k0 = 0; k0 < K; k0 += 4) {
                v2f av = *(const v2f*)(Ap + k0);
                if (reluA) { av.x = fmaxf(av.x, 0.f); av.y = fmaxf(av.y, 0.f); }
                v2f bv = *(const v2f*)(Bp + k0);
                acc = __builtin_amdgcn_wmma_f32_16x16x4_f32(
                    false, av, false, bv, (short)0, acc, false, false);
            }
        }
    } else {
        for (int k0 = 0; k0 < K; k0 += 4) {
            const int ka = k0 + 2 * half;
            float a0 = 0.f, a1 = 0.f;
            if (row < M) {
                if (ka     < K) a0 = A[(long long)row * lda + ka];
                if (ka + 1 < K) a1 = A[(long long)row * lda + ka + 1];
            }
            if (reluA) { a0 = fmaxf(a0, 0.f); a1 = fmaxf(a1, 0.f); }
            float b0 = 0.f, b1 = 0.f;
            if (col < N) {
                if (!transB) {
                    if (ka     < K) b0 = B[(long long)(ka    ) * ldb + col];
                    if (ka + 1 < K) b1 = B[(long long)(ka + 1) * ldb + col];
                } else {
                    if (ka     < K) b0 = B[(long long)col * ldb + ka];
                    if (ka + 1 < K) b1 = B[(long long)col * ldb + ka + 1];
                }
            }
            v2f av; av.x = a0; av.y = a1;
            v2f bv; bv.x = b0; bv.y = b1;
            acc = __builtin_amdgcn_wmma_f32_16x16x4_f32(
                false, av, false, bv, (short)0, acc, false, false);
        }
    }

    if (col < N) {
        const float bc = bias ? bias[col] : 0.f;
        for (int r = 0; r < 8; ++r) {
            const int rr = m0 + 8 * half + r;
            if (rr < M) {
                float v = acc[r] + bc;
                if (add)     v += add[(long long)rr * ldc + col];
                if (reluOut) v  = fmaxf(v, 0.f);
                C[(long long)rr * ldc + col] = v;
            }
        }
    }
}

// ---------------------------------------------------------------------------
// o_pair[i,h,c] = sum_j a[h,i,j] * LN(z)[i,j,c]   (LN fused via musd)
// Block = 128 threads = 4 waves; wave w handles 16-col tile (4*bx+w).
// M=12 (pad 16, head index clamped -> branchless loop), N=128, K=512.
// Writes into cat[i, 576 + h*128 + c].
// ---------------------------------------------------------------------------
__global__ void __launch_bounds__(128) opair_wmma(
                           const float* __restrict__ attn,
                           const float* __restrict__ z,
                           const float* __restrict__ musd,
                           const float* __restrict__ zg,
                           const float* __restrict__ zb,
                           float* __restrict__ cat)
{
    const int i    = blockIdx.z;
    const int wave = threadIdx.x >> 5;
    const int lane = threadIdx.x & 31;
    const int n0   = (blockIdx.x * 4 + wave) * 16;
    const int half = lane >> 4;
    const int lm   = lane & 15;
    const int h    = lm;                     // A row (head), rows 12..15 pad
    const int hA   = (h < HH) ? h : 0;       // clamp: padded rows compute on
                                             // head-0 data, never stored
    const int c    = n0 + lm;                // B/C col, always < 128

    const float g  = zg[c], bb = zb[c];
    const float* ar    = attn + (long long)hA * NRES * NRES + (long long)i * NRES
                              + 2 * half;
    const float* zbase = z    + (long long)i * NRES * CZ + c;
    const float* mbase = musd + (long long)i * NRES * 2 + 4 * half;

    v8f acc = {};
    for (int k0 = 0; k0 < NRES; k0 += 4) {
        const v2f av = *(const v2f*)(ar + k0);                 // a[h,i,j0..j0+1]
        const v4f m4 = *(const v4f*)(mbase + 2 * k0);          // mu0,rs0,mu1,rs1
        const int j0 = k0 + 2 * half;
        v2f bv;
        bv.x = (zbase[(long long)j0 * CZ]       - m4.x) * m4.y * g + bb;
        bv.y = (zbase[(long long)(j0 + 1) * CZ] - m4.z) * m4.w * g + bb;
        acc = __builtin_amdgcn_wmma_f32_16x16x4_f32(
            false, av, false, bv, (short)0, acc, false, false);
    }
    for (int r = 0; r < 8; ++r) {
        const int rr = 8 * half + r;         // head index
        if (rr < HH)
            cat[(long long)i * CATW + 576 + rr * CZ + c] = acc[r];
    }
}

// ---------------------------------------------------------------------------
// LayerNorm over last dim D (rows = gridDim.x), block = 128 threads.
// Safe in-place (values cached in registers before write).
// ---------------------------------------------------------------------------
__global__ void ln_kernel(const float* __restrict__ x,
                          const float* __restrict__ g,
                          const float* __restrict__ b,
                          float* __restrict__ out, int D)
{
    const int row = blockIdx.x;
    const float* xr = x + (long long)row * D;
    float* orow = out + (long long)row * D;
    __shared__ float red[128];
    const int tid = threadIdx.x;

    float vals[8];
    int cnt = 0;
    float s = 0.f;
    for (int c = tid; c < D; c += 128) { float v = xr[c]; vals[cnt++] = v; s += v; }
    red[tid] = s; __syncthreads();
    for (int st = 64; st > 0; st >>= 1) { if (tid < st) red[tid] += red[tid + st]; __syncthreads(); }
    const float mu = red[0] / (float)D; __syncthreads();

    float vs = 0.f;
    for (int k = 0; k < cnt; ++k) { float d = vals[k] - mu; vs += d * d; }
    red[tid] = vs; __syncthreads();
    for (int st = 64; st > 0; st >>= 1) { if (tid < st) red[tid] += red[tid + st]; __syncthreads(); }
    const float rs = rsqrtf(red[0] / (float)D + 1e-5f);

    cnt = 0;
    for (int c = tid; c < D; c += 128)
        orow[c] = (vals[cnt++] - mu) * rs * g[c] + b[c];
}

// Per-row mean / rsqrt(var+eps) for z (D = 128, block = 128).
__global__ void ln_stats_kernel(const float* __restrict__ x, float* __restrict__ musd)
{
    const long long row = blockIdx.x;
    __shared__ float red[128];
    const int tid = threadIdx.x;
    const float v = x[row * CZ + tid];
    red[tid] = v; __syncthreads();
    for (int st = 64; st > 0; st >>= 1) { if (tid < st) red[tid] += red[tid + st]; __syncthreads(); }
    const float mu = red[0] / (float)CZ; __syncthreads();
    const float d = v - mu;
    red[tid] = d * d; __syncthreads();
    for (int st = 64; st > 0; st >>= 1) { if (tid < st) red[tid] += red[tid + st]; __syncthreads(); }
    if (tid == 0) {
        musd[row * 2]     = mu;
        musd[row * 2 + 1] = rsqrtf(red[0] / (float)CZ + 1e-5f);
    }
}

// biasm[i,j,h] = sqrt(1/3)*(LN(z)[i,j]·wb[:,h] + bb[h]) + (m_i*m_j - 1)*INF
__global__ void bias_kernel(const float* __restrict__ z,
                            const float* __restrict__ musd,
                            const float* __restrict__ zg,
                            const float* __restrict__ zb,
                            const float* __restrict__ wb,
                            const float* __restrict__ bb,
                            const float* __restrict__ mask,
                            float* __restrict__ biasm)
{
    const long long row = blockIdx.x;            // i*N + j
    const int i = (int)(row / NRES), j = (int)(row % NRES);
    const int tid = threadIdx.x;                 // 128
    const float mu = musd[row * 2], rs = musd[row * 2 + 1];
    const float zv = (z[row * CZ + tid] - mu) * rs * zg[tid] + zb[tid];
    __shared__ float p[128][HH + 1];
    for (int h = 0; h < HH; ++h) p[tid][h] = zv * wb[tid * HH + h];
    __syncthreads();
    if (tid < HH) {
        float s = 0.f;
        for (int c = 0; c < 128; ++c) s += p[c][tid];
        const float mt = (mask[i] * mask[j] - 1.f) * 100000.f;
        biasm[row * HH + tid] = 0.57735026919f * (s + bb[tid]) + mt;
    }
}

// Build Q/K attention feature vectors (K=32) and [v | v_pts] (width 40).
__global__ void featprep_kernel(const float* __restrict__ q_raw,
                                const float* __restrict__ kv_raw,
                                const float* __restrict__ qp_raw,
                                const float* __restrict__ kvp_raw,
                                const float* __restrict__ head_w,
                                const float* __restrict__ R,
                                const float* __restrict__ t,
                                float* __restrict__ Qf,
                                float* __restrict__ Kf,
                                float* __restrict__ vcat)
{
    const int n = blockIdx.x;
    const int h = threadIdx.x;      // block = 16
    if (h >= HH) return;

    float Rm[9], tv[3];
    for (int k = 0; k < 9; ++k) Rm[k] = R[n * 9 + k];
    for (int k = 0; k < 3; ++k) tv[k] = t[n * 3 + k];

    const float hw = log1pf(expf(head_w[h])) * rsqrtf(54.0f); // softplus * sqrt(1/(3*PQ*4.5))
    const float c1 = rsqrtf(48.0f);                           // sqrt(1/(3*CI))

    float* qf = Qf   + ((long long)n * HH + h) * 32;
    float* kf = Kf   + ((long long)n * HH + h) * 32;
    float* vc = vcat + ((long long)n * HH + h) * 40;

    for (int c = 0; c < CI; ++c) {
        qf[c] = c1 * q_raw[(long long)n * (HH * CI) + h * CI + c];
        kf[c] = kv_raw[(long long)n * (HH * 2 * CI) + h * 2 * CI + c];
        vc[c] = kv_raw[(long long)n * (HH * 2 * CI) + h * 2 * CI + CI + c];
    }
    float qss = 0.f;
    for (int p = 0; p < PQ; ++p) {
        const float* lp = qp_raw + (long long)n * (HH * PQ * 3) + (h * PQ + p) * 3;
        const float gx = Rm[0]*lp[0] + Rm[1]*lp[1] + Rm[2]*lp[2] + tv[0];
        const float gy = Rm[3]*lp[0] + Rm[4]*lp[1] + Rm[5]*lp[2] + tv[1];
        const float gz = Rm[6]*lp[0] + Rm[7]*lp[1] + Rm[8]*lp[2] + tv[2];
        qf[16 + p * 3 + 0] = hw * gx;
        qf[16 + p * 3 + 1] = hw * gy;
        qf[16 + p * 3 + 2] = hw * gz;
        qss += gx * gx + gy * gy + gz * gz;
    }
    qf[28] = -0.5f * hw * qss; qf[29] = 1.f; qf[30] = 0.f; qf[31] = 0.f;

    float kss = 0.f;
    for (int p = 0; p < PQ + PV; ++p) {
        const float* lp = kvp_raw + (long long)n * (HH * (PQ + PV) * 3) + (h * (PQ + PV) + p) * 3;
        const float gx = Rm[0]*lp[0] + Rm[1]*lp[1] + Rm[2]*lp[2] + tv[0];
        const float gy = Rm[3]*lp[0] + Rm[4]*lp[1] + Rm[5]*lp[2] + tv[1];
        const float gz = Rm[6]*lp[0] + Rm[7]*lp[1] + Rm[8]*lp[2] + tv[2];
        if (p < PQ) {
            kf[16 + p * 3 + 0] = gx;
            kf[16 + p * 3 + 1] = gy;
            kf[16 + p * 3 + 2] = gz;
            kss += gx * gx + gy * gy + gz * gz;
        } else {
            const int pv = p - PQ;
            vc[16 + pv * 3 + 0] = gx;
            vc[16 + pv * 3 + 1] = gy;
            vc[16 + pv * 3 + 2] = gz;
        }
    }
    kf[28] = 1.f; kf[29] = -0.5f * hw * kss; kf[30] = 0.f; kf[31] = 0.f;
}

// Softmax over j (row length 512) with pair-bias + mask added.
__global__ void softmax_kernel(float* __restrict__ attn, const float* __restrict__ biasm)
{
    const int blk = blockIdx.x;           // h*N + i
    const int h = blk / NRES, i = blk % NRES;
    float* row = attn + (long long)h * NRES * NRES + (long long)i * NRES;
    const float* bm = biasm + (long long)i * NRES * HH;
    const int tid = threadIdx.x;          // 256
    float v0 = row[tid]       + bm[(long long)tid * HH + h];
    float v1 = row[tid + 256] + bm[(long long)(tid + 256) * HH + h];
    __shared__ float red[256];
    red[tid] = fmaxf(v0, v1); __syncthreads();
    for (int st = 128; st > 0; st >>= 1) { if (tid < st) red[tid] = fmaxf(red[tid], red[tid + st]); __syncthreads(); }
    const float m = red[0]; __syncthreads();
    const float e0 = expf(v0 - m), e1 = expf(v1 - m);
    red[tid] = e0 + e1; __syncthreads();
    for (int st = 128; st > 0; st >>= 1) { if (tid < st) red[tid] += red[tid + st]; __syncthreads(); }
    const float inv = 1.f / red[0];
    row[tid] = e0 * inv; row[tid + 256] = e1 * inv;
}

// Inverse-rigid the attended points, norms, and copy o into cat.
__global__ void finalize_kernel(const float* __restrict__ ov,
                                const float* __restrict__ R,
                                const float* __restrict__ t,
                                float* __restrict__ cat)
{
    const int i = blockIdx.x;
    const int tid = threadIdx.x;          // 96
    const long long base = (long long)i * CATW;
    for (int idx = tid; idx < HH * CI; idx += 96) {
        const int h = idx >> 4, c = idx & 15;
        cat[base + idx] = ov[((long long)i * HH + h) * 40 + c];
    }
    const int h = tid / PV, p = tid % PV;
    const float* g = ov + ((long long)i * HH + h) * 40 + 16 + p * 3;
    const float d0 = g[0] - t[i * 3], d1 = g[1] - t[i * 3 + 1], d2 = g[2] - t[i * 3 + 2];
    const float* Rm = R + i * 9;
    const float lx = Rm[0]*d0 + Rm[3]*d1 + Rm[6]*d2;   // R^T * d
    const float ly = Rm[1]*d0 + Rm[4]*d1 + Rm[7]*d2;
    const float lz = Rm[2]*d0 + Rm[5]*d1 + Rm[8]*d2;
    cat[base + 192 + h * PV + p] = lx;
    cat[base + 288 + h * PV + p] = ly;
    cat[base + 384 + h * PV + p] = lz;
    cat[base + 480 + h * PV + p] = sqrtf(lx*lx + ly*ly + lz*lz + 1e-8f);
}

__global__ void init_rt_kernel(float* __restrict__ R, float* __restrict__ t)
{
    const int n = blockIdx.x * blockDim.x + threadIdx.x;
    if (n >= NRES) return;
    for (int k = 0; k < 9; ++k) R[n * 9 + k] = (k % 4 == 0) ? 1.f : 0.f;
    t[n * 3] = 0.f; t[n * 3 + 1] = 0.f; t[n * 3 + 2] = 0.f;
}

// Backbone update: quat -> R_u; t += R*v; R = R @ R_u.
__global__ void bbupd_kernel(const float* __restrict__ upd,
                             float* __restrict__ R, float* __restrict__ t)
{
    const int n = blockIdx.x * blockDim.x + threadIdx.x;
    if (n >= NRES) return;
    const float u0 = upd[n*6], u1 = upd[n*6+1], u2 = upd[n*6+2];
    const float inv = rsqrtf(1.f + u0*u0 + u1*u1 + u2*u2);
    const float w = inv, x = u0*inv, y = u1*inv, z = u2*inv;
    const float Ru[9] = {
        1-2*(y*y+z*z), 2*(x*y-w*z),   2*(x*z+w*y),
        2*(x*y+w*z),   1-2*(x*x+z*z), 2*(y*z-w*x),
        2*(x*z-w*y),   2*(y*z+w*x),   1-2*(x*x+y*y) };
    float Rm[9];
    for (int k = 0; k < 9; ++k) Rm[k] = R[n*9+k];
    const float v0 = upd[n*6+3], v1 = upd[n*6+4], v2 = upd[n*6+5];
    t[n*3+0] += Rm[0]*v0 + Rm[1]*v1 + Rm[2]*v2;
    t[n*3+1] += Rm[3]*v0 + Rm[4]*v1 + Rm[5]*v2;
    t[n*3+2] += Rm[6]*v0 + Rm[7]*v1 + Rm[8]*v2;
    for (int r = 0; r < 3; ++r)
        for (int c = 0; c < 3; ++c)
            R[n*9 + r*3 + c] = Rm[r*3+0]*Ru[0*3+c] + Rm[r*3+1]*Ru[1*3+c] + Rm[r*3+2]*Ru[2*3+c];
}

// out[i] = [ s (384) | t*10 (3) | angles (14) | unnorm_angles (14) ]
__global__ void output_kernel(const float* __restrict__ s,
                              const float* __restrict__ t,
                              const float* __restrict__ ang,
                              float* __restrict__ out)
{
    const int i = blockIdx.x;
    const int tid = threadIdx.x;          // 128
    const long long ob = (long long)i * OUTW;
    for (int c = tid; c < CS; c += 128) out[ob + c] = s[(long long)i * CS + c];
    if (tid < 3) out[ob + CS + tid] = t[i * 3 + tid] * 10.f;
    if (tid < 7) {
        const float a0 = ang[i * 14 + tid * 2], a1 = ang[i * 14 + tid * 2 + 1];
        const float dn = rsqrtf(fmaxf(a0 * a0 + a1 * a1, 1e-8f));
        out[ob + 387 + tid * 2]     = a0 * dn;
        out[ob + 387 + tid * 2 + 1] = a1 * dn;
        out[ob + 401 + tid * 2]     = a0;
        out[ob + 401 + tid * 2 + 1] = a1;
    }
}

// ---------------------------------------------------------------------------
// Host side
// ---------------------------------------------------------------------------
static inline void gemm(hipStream_t st, const float* A, const float* B,
                        const float* bias, const float* add, float* C,
                        int M, int N, int K, int lda, int ldb, int ldc,
                        long long sA, long long sB, long long sC,
                        int batch, int transB, int reluA, int reluOut)
{
    dim3 grid((N + 63) / 64, (M + 15) / 16, batch);
    gemm_wmma_f32<<<grid, 128, 0, st>>>(A, B, bias, add, C, M, N, K,
                                        lda, ldb, ldc, sA, sB, sC,
                                        transB, reluA, reluOut);
}

// Input indices: harness flattens the pytree JAX-style (dict keys sorted).
enum {
    IN_MASK = 0,
    IN_ANG_B_IN, IN_ANG_B_INIT, IN_ANG_B_OUT,
    IN_ANG_BLK0_B1, IN_ANG_BLK0_B2, IN_ANG_BLK0_W1, IN_ANG_BLK0_W2,
    IN_ANG_BLK1_B1, IN_ANG_BLK1_B2, IN_ANG_BLK1_W1, IN_ANG_BLK1_W2,
    IN_ANG_W_IN, IN_ANG_W_INIT, IN_ANG_W_OUT,
    IN_B_BB, IN_B_IN,
    IN_IPA_BB, IN_IPA_BKV, IN_IPA_BKVP, IN_IPA_BO, IN_IPA_BQ, IN_IPA_BQP,
    IN_IPA_HEAD_W, IN_IPA_WB, IN_IPA_WKV, IN_IPA_WKVP, IN_IPA_WO, IN_IPA_WQ, IN_IPA_WQP,
    IN_LN_IPA_B, IN_LN_IPA_G, IN_LN_S_B, IN_LN_S_G, IN_LN_Z_B, IN_LN_Z_G,
    IN_TR_B1, IN_TR_B2, IN_TR_B3, IN_TR_LN_B, IN_TR_LN_G, IN_TR_W1, IN_TR_W2, IN_TR_W3,
    IN_W_BB, IN_W_IN,
    IN_S, IN_Z
};

extern "C" void kernel_launch(void* const* d_in, const int* in_sizes, int n_in,
                              void* d_out, int out_size, void* d_ws, size_t ws_size,
                              hipStream_t stream)
{
    (void)in_sizes; (void)n_in; (void)out_size; (void)ws_size;
    const int N = NRES;
    #define F(idx) ((const float*)d_in[idx])

    // ---- workspace carve (deterministic) ----
    float* p = (float*)d_ws;
    auto alloc = [&](long long n) { float* r = p; p += n; return r; };
    float* s_init  = alloc((long long)N * CS);
    float* s_cur   = alloc((long long)N * CS);
    float* musd    = alloc((long long)N * N * 2);
    float* biasm   = alloc((long long)N * N * HH);
    float* attn    = alloc((long long)HH * N * N);
    float* q_raw   = alloc((long long)N * HH * CI);
    float* kv_raw  = alloc((long long)N * HH * 2 * CI);
    float* qp_raw  = alloc((long long)N * HH * PQ * 3);
    float* kvp_raw = alloc((long long)N * HH * (PQ + PV) * 3);
    float* Qf      = alloc((long long)N * HH * 32);
    float* Kf      = alloc((long long)N * HH * 32);
    float* vcat    = alloc((long long)N * HH * 40);
    float* ov      = alloc((long long)N * HH * 40);
    float* cat     = alloc((long long)N * CATW);
    float* Rb      = alloc((long long)N * 9);
    float* tb      = alloc((long long)N * 3);
    float* updb    = alloc((long long)N * 6);
    float* t1      = alloc((long long)N * CS);
    float* t2      = alloc((long long)N * CS);
    float* a0      = alloc((long long)N * CR);
    float* atmp    = alloc((long long)N * CR);
    float* ang     = alloc((long long)N * 14);

    // ---- invariant precompute ----
    ln_kernel<<<N, 128, 0, stream>>>(F(IN_S), F(IN_LN_S_G), F(IN_LN_S_B), s_init, CS);
    gemm(stream, s_init, F(IN_W_IN), F(IN_B_IN), nullptr, s_cur,
         N, CS, CS, CS, CS, CS, 0, 0, 0, 1, 0, 0, 0);
    ln_stats_kernel<<<N * N, 128, 0, stream>>>(F(IN_Z), musd);
    bias_kernel<<<N * N, 128, 0, stream>>>(F(IN_Z), musd, F(IN_LN_Z_G), F(IN_LN_Z_B),
                                           F(IN_IPA_WB), F(IN_IPA_BB), F(IN_MASK), biasm);
    init_rt_kernel<<<2, 256, 0, stream>>>(Rb, tb);

    // ---- 8 structure blocks ----
    for (int blk = 0; blk < 8; ++blk) {
        // projections from s
        gemm(stream, s_cur, F(IN_IPA_WQ),  F(IN_IPA_BQ),  nullptr, q_raw,
             N, HH * CI, CS, CS, HH * CI, HH * CI, 0, 0, 0, 1, 0, 0, 0);
        gemm(stream, s_cur, F(IN_IPA_WKV), F(IN_IPA_BKV), nullptr, kv_raw,
             N, HH * 2 * CI, CS, CS, HH * 2 * CI, HH * 2 * CI, 0, 0, 0, 1, 0, 0, 0);
        gemm(stream, s_cur, F(IN_IPA_WQP), F(IN_IPA_BQP), nullptr, qp_raw,
             N, HH * PQ * 3, CS, CS, HH * PQ * 3, HH * PQ * 3, 0, 0, 0, 1, 0, 0, 0);
        gemm(stream, s_cur, F(IN_IPA_WKVP), F(IN_IPA_BKVP), nullptr, kvp_raw,
             N, HH * (PQ + PV) * 3, CS, CS, HH * (PQ + PV) * 3, HH * (PQ + PV) * 3,
             0, 0, 0, 1, 0, 0, 0);

        featprep_kernel<<<N, 16, 0, stream>>>(q_raw, kv_raw, qp_raw, kvp_raw,
                                              F(IN_IPA_HEAD_W), Rb, tb, Qf, Kf, vcat);

        // attention logits: batched over heads, Qf @ Kf^T, K=32
        gemm(stream, Qf, Kf, nullptr, nullptr, attn,
             N, N, 32, HH * 32, HH * 32, N, 32, 32, (long long)N * N,
             HH, /*transB=*/1, 0, 0);
        softmax_kernel<<<HH * N, 256, 0, stream>>>(attn, biasm);

        // attended values + points: batched over heads, a @ [v|v_pts]
        gemm(stream, attn, vcat, nullptr, nullptr, ov,
             N, 40, N, N, HH * 40, HH * 40, (long long)N * N, 40, 40,
             HH, 0, 0, 0);

        // o_pair with fused LN(z)
        {
            dim3 grid(CZ / 64, 1, N);
            opair_wmma<<<grid, 128, 0, stream>>>(attn, F(IN_Z), musd,
                                                 F(IN_LN_Z_G), F(IN_LN_Z_B), cat);
        }
        finalize_kernel<<<N, 96, 0, stream>>>(ov, Rb, tb, cat);

        // s = s + cat @ wo + bo ; LN
        gemm(stream, cat, F(IN_IPA_WO), F(IN_IPA_BO), s_cur, s_cur,
             N, CS, CATW, CATW, CS, CS, 0, 0, 0, 1, 0, 0, 0);
        ln_kernel<<<N, 128, 0, stream>>>(s_cur, F(IN_LN_IPA_G), F(IN_LN_IPA_B), s_cur, CS);

        // transition
        gemm(stream, s_cur, F(IN_TR_W1), F(IN_TR_B1), nullptr, t1,
             N, CS, CS, CS, CS, CS, 0, 0, 0, 1, 0, 0, 1);
        gemm(stream, t1, F(IN_TR_W2), F(IN_TR_B2), nullptr, t2,
             N, CS, CS, CS, CS, CS, 0, 0, 0, 1, 0, 0, 1);
        gemm(stream, t2, F(IN_TR_W3), F(IN_TR_B3), s_cur, s_cur,
             N, CS, CS, CS, CS, CS, 0, 0, 0, 1, 0, 0, 0);
        ln_kernel<<<N, 128, 0, stream>>>(s_cur, F(IN_TR_LN_G), F(IN_TR_LN_B), s_cur, CS);

        // backbone update
        gemm(stream, s_cur, F(IN_W_BB), F(IN_B_BB), nullptr, updb,
             N, 6, CS, CS, 6, 6, 0, 0, 0, 1, 0, 0, 0);
        bbupd_kernel<<<2, 256, 0, stream>>>(updb, Rb, tb);
    }

    // ---- angle resnet (only final iteration's angles are observable) ----
    gemm(stream, s_cur, F(IN_ANG_W_IN), F(IN_ANG_B_IN), nullptr, a0,
         N, CR, CS, CS, CR, CR, 0, 0, 0, 1, 0, 1, 0);
    gemm(stream, s_init, F(IN_ANG_W_INIT), F(IN_ANG_B_INIT), a0, a0,
         N, CR, CS, CS, CR, CR, 0, 0, 0, 1, 0, 1, 0);
    const float* bw1[2] = { F(IN_ANG_BLK0_W1), F(IN_ANG_BLK1_W1) };
    const float* bw2[2] = { F(IN_ANG_BLK0_W2), F(IN_ANG_BLK1_W2) };
    const float* bb1[2] = { F(IN_ANG_BLK0_B1), F(IN_ANG_BLK1_B1) };
    const float* bb2[2] = { F(IN_ANG_BLK0_B2), F(IN_ANG_BLK1_B2) };
    for (int r = 0; r < 2; ++r) {
        gemm(stream, a0, bw1[r], bb1[r], nullptr, atmp,
             N, CR, CR, CR, CR, CR, 0, 0, 0, 1, 0, 1, 0);
        gemm(stream, atmp, bw2[r], bb2[r], a0, a0,
             N, CR, CR, CR, CR, CR, 0, 0, 0, 1, 0, 1, 0);
    }
    gemm(stream, a0, F(IN_ANG_W_OUT), F(IN_ANG_B_OUT), nullptr, ang,
         N, 14, CR, CR, 14, 14, 0, 0, 0, 1, 0, 1, 0);

    output_kernel<<<N, 128, 0, stream>>>(s_cur, tb, ang, (float*)d_out);
    #undef F
}